// GraphConvolution_6966436954119
// MI455X (gfx1250) — compile-verified
//
#include <hip/hip_runtime.h>
#include <hip/hip_bf16.h>

typedef __attribute__((ext_vector_type(2))) float v2f;
typedef __attribute__((ext_vector_type(8))) float v8f;

#define F_IN  256
#define F_OUT 128
// K-pair-interleaved W row: 2*F_OUT floats + 32-dword skew so the two
// half-waves (kp vs kp+1, 288 mod 64 = 32) hit disjoint LDS bank windows.
#define PKROW 288

// ---------------------------------------------------------------------------
// Kernel 1: zero the output accumulator (float4-wide)
// ---------------------------------------------------------------------------
__global__ void __launch_bounds__(256) gc_zero_kernel(float4* __restrict__ out,
                                                      int total4) {
    int i = blockIdx.x * 256 + threadIdx.x;
    if (i < total4) out[i] = make_float4(0.f, 0.f, 0.f, 0.f);
}

// ---------------------------------------------------------------------------
// Kernel 2: pre_sup[N, 128] = x[N, 256] @ W[256, 128] via V_WMMA_F32_16X16X4_F32
// Block = 256 threads = 8 waves; each wave computes a 16-row x 128-col strip.
// W staged in LDS in K-pair-interleaved layout (144 KB of the 320 KB WGP LDS):
//   Wl[kp*PKROW + 2*n + 0] = W[2*kp    ][n]
//   Wl[kp*PKROW + 2*n + 1] = W[2*kp + 1][n]
// so a B fragment (W[kk][n], W[kk+1][n]) is ONE ds_load_b64 straight into an
// even-aligned VGPR pair (no v_mov repacking, no bank conflicts).
// A-frag (16x4 f32): lanes 0-15 -> K=k,k+1; lanes 16-31 -> K=k+2,k+3 (b64 load).
// C/D (16x16 f32): VGPR r, lanes 0-15 -> M=r; lanes 16-31 -> M=8+r.
// ---------------------------------------------------------------------------
__global__ void __launch_bounds__(256) gc_gemm_kernel(const float* __restrict__ x,
                                                      const float* __restrict__ W,
                                                      float* __restrict__ pre,
                                                      int n_nodes) {
    __shared__ float Wl[(F_IN / 2) * PKROW];  // 144 KB LDS

    const int tid = threadIdx.x;
    // Pack W -> LDS: work item = (kp, 4-column group); b128 loads + b128 stores.
    for (int i = tid; i < (F_IN / 2) * (F_OUT / 4); i += 256) {
        const int kp = i >> 5;      // 0..127  (k-pair)
        const int nq = i & 31;      // 0..31   (columns 4*nq .. 4*nq+3)
        const float4 r0 = *(const float4*)(W + (2 * kp)     * F_OUT + 4 * nq);
        const float4 r1 = *(const float4*)(W + (2 * kp + 1) * F_OUT + 4 * nq);
        float4* dst = (float4*)(Wl + kp * PKROW + 8 * nq);
        dst[0] = make_float4(r0.x, r1.x, r0.y, r1.y);
        dst[1] = make_float4(r0.z, r1.z, r0.w, r1.w);
    }
    __syncthreads();

    const int wave  = tid >> 5;      // 0..7
    const int lane  = tid & 31;
    const int lhalf = lane >> 4;     // 0 or 1
    const int l16   = lane & 15;

    const int row_base = (blockIdx.x * 8 + wave) * 16;
    if (row_base >= n_nodes) return;   // uniform per wave: EXEC stays all-1s

    v8f acc[8];
    #pragma unroll
    for (int t = 0; t < 8; ++t) acc[t] = v8f{};

    // Clamp A row for partial tiles (stores are guarded below)
    int arow = row_base + l16;
    if (arow >= n_nodes) arow = n_nodes - 1;
    const float* __restrict__ xrow = x + (size_t)arow * F_IN;

    for (int k = 0; k < F_IN; k += 4) {
        // A fragment: two consecutive K values for this lane -> global_load_b64
        const v2f a = *(const v2f*)(xrow + k + 2 * lhalf);
        const float* __restrict__ brow = Wl + ((k >> 1) + lhalf) * PKROW;
        #pragma unroll
        for (int t = 0; t < 8; ++t) {
            // B fragment: one ds_load_b64 into an even-aligned pair
            const v2f b = *(const v2f*)(brow + 2 * (t * 16 + l16));
            acc[t] = __builtin_amdgcn_wmma_f32_16x16x4_f32(
                false, a, false, b, (short)0, acc[t], false, false);
        }
    }

    // Store D: VGPR r -> row = row_base + r + 8*lhalf, col = t*16 + l16
    #pragma unroll
    for (int t = 0; t < 8; ++t) {
        #pragma unroll
        for (int r = 0; r < 8; ++r) {
            const int rr = row_base + r + 8 * lhalf;
            if (rr < n_nodes)
                pre[(size_t)rr * F_OUT + t * 16 + l16] = acc[t][r];
        }
    }
}

// ---------------------------------------------------------------------------
// Kernel 3: SpMM scatter: out[row[e],:] += vals[e] * pre[col[e],:]
// One wave (32 lanes) per edge; each lane: float4 gather + 4 f32 atomics.
// out + pre (~102 MB) are resident in the 192 MB L2, so the scattered
// atomics resolve at the L2 atomic units without HBM round trips.
// ---------------------------------------------------------------------------
__global__ void __launch_bounds__(256) gc_spmm_kernel(const float* __restrict__ pre,
                                                      const float* __restrict__ vals,
                                                      const int* __restrict__ row,
                                                      const int* __restrict__ col,
                                                      float* __restrict__ out,
                                                      int n_edges) {
    const int e = blockIdx.x * 8 + (threadIdx.x >> 5);
    if (e >= n_edges) return;
    const int lane = threadIdx.x & 31;

    const int   r = row[e];
    const int   c = col[e];
    const float v = vals[e];

    const float4 m = ((const float4*)(pre + (size_t)c * F_OUT))[lane];
    float* op = out + (size_t)r * F_OUT + lane * 4;
    atomicAdd(op + 0, v * m.x);
    atomicAdd(op + 1, v * m.y);
    atomicAdd(op + 2, v * m.z);
    atomicAdd(op + 3, v * m.w);
}

// ---------------------------------------------------------------------------
// Kernel 4: out = relu(out + b), float4-wide
// ---------------------------------------------------------------------------
__global__ void __launch_bounds__(256) gc_bias_relu_kernel(float4* __restrict__ out,
                                                           const float4* __restrict__ b,
                                                           int total4) {
    int i = blockIdx.x * 256 + threadIdx.x;
    if (i < total4) {
        const float4 o  = out[i];
        const float4 bb = b[i & (F_OUT / 4 - 1)];
        float4 v;
        v.x = o.x + bb.x; v.y = o.y + bb.y;
        v.z = o.z + bb.z; v.w = o.w + bb.w;
        v.x = v.x > 0.f ? v.x : 0.f;
        v.y = v.y > 0.f ? v.y : 0.f;
        v.z = v.z > 0.f ? v.z : 0.f;
        v.w = v.w > 0.f ? v.w : 0.f;
        out[i] = v;
    }
}

// ---------------------------------------------------------------------------
extern "C" void kernel_launch(void* const* d_in, const int* in_sizes, int n_in,
                              void* d_out, int out_size, void* d_ws, size_t ws_size,
                              hipStream_t stream) {
    const float* x    = (const float*)d_in[0];
    const float* W    = (const float*)d_in[1];
    const float* bias = (const float*)d_in[2];
    const float* vals = (const float*)d_in[3];
    const int*   row  = (const int*)d_in[4];
    const int*   col  = (const int*)d_in[5];
    float*       out  = (float*)d_out;

    const int n_nodes = in_sizes[0] / F_IN;
    const int n_edges = in_sizes[3];
    const int total   = n_nodes * F_OUT;
    const int total4  = total / 4;

    float* pre = (float*)d_ws;  // needs n_nodes * 128 * 4 = 51.2 MB scratch

    // 1) zero output accumulator
    gc_zero_kernel<<<(total4 + 255) / 256, 256, 0, stream>>>((float4*)out, total4);

    // 2) dense projection via WMMA (128 rows per block: 8 waves x 16 rows)
    const int gemm_blocks = (n_nodes + 127) / 128;
    gc_gemm_kernel<<<gemm_blocks, 256, 0, stream>>>(x, W, pre, n_nodes);

    // 3) edge scatter-add (8 edges per block, one wave each)
    const int spmm_blocks = (n_edges + 7) / 8;
    gc_spmm_kernel<<<spmm_blocks, 256, 0, stream>>>(pre, vals, row, col, out, n_edges);

    // 4) bias + relu
    gc_bias_relu_kernel<<<(total4 + 255) / 256, 256, 0, stream>>>(
        (float4*)out, (const float4*)bias, total4);
}